// MaskGen_34522947125786
// MI455X (gfx1250) — compile-verified
//
#include <hip/hip_runtime.h>

typedef float v2f __attribute__((ext_vector_type(2)));
typedef float v8f __attribute__((ext_vector_type(8)));

#define HW_SZ          25600            // 160*160
#define C_SZ           64
#define B_SZ           32
#define PIX_PER_BLOCK  128              // 8 waves * 16 pixels
#define BLOCKS_PER_IMG (HW_SZ / PIX_PER_BLOCK)   // 200
#define NBLK1          (B_SZ * BLOCKS_PER_IMG)   // 6400
#define NTOT           (B_SZ * HW_SZ)            // 819200
#define BN_EPS_F       1e-5f
#define SLOPE_F        0.1f

// Phase 1: per-image channel GEMV via V_WMMA_F32_16X16X4_F32, plus per-block
// partial sums (sum, sumsq) for the global BatchNorm statistics.
__global__ __launch_bounds__(256) void maskgen_wmma(
    const float* __restrict__ sf, const float* __restrict__ feats,
    float* __restrict__ out, float* __restrict__ partials)
{
    const int lane  = threadIdx.x & 31;
    const int wave  = threadIdx.x >> 5;
    const int img   = blockIdx.x / BLOCKS_PER_IMG;
    const int chunk = blockIdx.x % BLOCKS_PER_IMG;
    const int p0    = chunk * PIX_PER_BLOCK + wave * 16;

    const float* __restrict__ sfb = sf + img * C_SZ;      // uniform -> s_loads
    const float* __restrict__ fp  =
        feats + (size_t)img * C_SZ * HW_SZ + p0 + (lane & 15);

    const int hi = lane >> 4;        // 0: lanes 0-15, 1: lanes 16-31
    const int m  = lane & 15;        // pixel-in-half / column id

    v8f acc = {};
    #pragma unroll
    for (int c = 0; c < C_SZ; c += 4) {
        // A (16x4 f32): vgpr j, half hi -> K = c + 2*hi + j
        const int ch = c + (hi << 1);
        v2f a, bm;
        a[0] = fp[(size_t)(ch + 0) * HW_SZ];
        a[1] = fp[(size_t)(ch + 1) * HW_SZ];
        // B (4x16 f32, broadcast along N): vgpr0 = K0|K2, vgpr1 = K1|K3
        const float s0 = sfb[c + 0], s1 = sfb[c + 1];
        const float s2 = sfb[c + 2], s3 = sfb[c + 3];
        bm[0] = hi ? s2 : s0;
        bm[1] = hi ? s3 : s1;
        acc = __builtin_amdgcn_wmma_f32_16x16x4_f32(
                  false, a, false, bm, (short)0, acc, false, false);
    }

    // D layout: lane (16*hi + n), vgpr v -> D[M = v + 8*hi][n]; all columns equal.
    // Store the 16 mask values: lanes with n == v write pixel p0 + v + 8*hi.
    float* __restrict__ op = out + (size_t)img * HW_SZ + p0;
    if (m < 8) {
        float val = acc[0];
        #pragma unroll
        for (int v = 1; v < 8; ++v) if (m == v) val = acc[v];
        op[m + (hi << 3)] = val;
    }

    // Per-lane column sums / sumsq; partner (xor 16) holds the other 8 rows.
    float ps = acc[0] + acc[1] + acc[2] + acc[3] + acc[4] + acc[5] + acc[6] + acc[7];
    float qs = acc[0]*acc[0] + acc[1]*acc[1] + acc[2]*acc[2] + acc[3]*acc[3]
             + acc[4]*acc[4] + acc[5]*acc[5] + acc[6]*acc[6] + acc[7]*acc[7];
    float fullS = ps + __shfl_xor(ps, 16, 32);
    float fullQ = qs + __shfl_xor(qs, 16, 32);

    __shared__ float sS[8], sQ[8];
    if (lane == 0) { sS[wave] = fullS; sQ[wave] = fullQ; }
    __syncthreads();
    if (threadIdx.x == 0) {
        float S = 0.f, Q = 0.f;
        #pragma unroll
        for (int w = 0; w < 8; ++w) { S += sS[w]; Q += sQ[w]; }
        partials[2 * blockIdx.x + 0] = S;
        partials[2 * blockIdx.x + 1] = Q;
    }
}

// Phase 2: deterministic single-block reduction of 6400 (sum, sumsq) pairs.
__global__ __launch_bounds__(256) void reduce_stats(
    const float* __restrict__ partials, float* __restrict__ stats)
{
    __shared__ float sS[256], sQ[256];
    float S = 0.f, Q = 0.f;
    for (int i = threadIdx.x; i < NBLK1; i += 256) {
        S += partials[2 * i + 0];
        Q += partials[2 * i + 1];
    }
    sS[threadIdx.x] = S; sQ[threadIdx.x] = Q;
    __syncthreads();
    for (int s = 128; s > 0; s >>= 1) {
        if ((int)threadIdx.x < s) {
            sS[threadIdx.x] += sS[threadIdx.x + s];
            sQ[threadIdx.x] += sQ[threadIdx.x + s];
        }
        __syncthreads();
    }
    if (threadIdx.x == 0) {
        const float invN = 1.0f / (float)NTOT;
        const float mean = sS[0] * invN;
        const float var  = sQ[0] * invN - mean * mean;   // biased variance
        stats[0] = mean;
        stats[1] = rsqrtf(var + BN_EPS_F);
    }
}

// Phase 3: normalize + affine + LeakyReLU, float4-vectorized, in place.
__global__ __launch_bounds__(256) void normalize_lrelu(
    float* __restrict__ out, const float* __restrict__ stats,
    const float* __restrict__ bn_w, const float* __restrict__ bn_b)
{
    const float mean = stats[0];
    const float g    = bn_w[0] * stats[1];        // weight * inv_std
    const float bb   = bn_b[0] - mean * g;        // y = x*g + bb
    const int i = blockIdx.x * blockDim.x + threadIdx.x;   // NTOT/4 threads
    float4* o4 = (float4*)out;
    float4 v = o4[i];
    float y0 = v.x * g + bb, y1 = v.y * g + bb;
    float y2 = v.z * g + bb, y3 = v.w * g + bb;
    v.x = (y0 >= 0.f) ? y0 : SLOPE_F * y0;
    v.y = (y1 >= 0.f) ? y1 : SLOPE_F * y1;
    v.z = (y2 >= 0.f) ? y2 : SLOPE_F * y2;
    v.w = (y3 >= 0.f) ? y3 : SLOPE_F * y3;
    o4[i] = v;
}

extern "C" void kernel_launch(void* const* d_in, const int* in_sizes, int n_in,
                              void* d_out, int out_size, void* d_ws, size_t ws_size,
                              hipStream_t stream) {
    const float* sf    = (const float*)d_in[0];   // [32,64,1,1]
    const float* feats = (const float*)d_in[1];   // [32,64,160,160]
    const float* bn_w  = (const float*)d_in[2];   // [1]
    const float* bn_b  = (const float*)d_in[3];   // [1]
    float* out      = (float*)d_out;              // [32,1,160,160]
    float* partials = (float*)d_ws;               // 2*NBLK1 floats
    float* stats    = partials + 2 * NBLK1;       // 2 floats

    maskgen_wmma   <<<NBLK1, 256, 0, stream>>>(sf, feats, out, partials);
    reduce_stats   <<<1, 256, 0, stream>>>(partials, stats);
    normalize_lrelu<<<NTOT / 4 / 256, 256, 0, stream>>>(out, stats, bn_w, bn_b);
}